// SingleMessagePassing_1623497638281
// MI455X (gfx1250) — compile-verified
//
#include <hip/hip_runtime.h>
#include <hip/hip_bf16.h>

// ---------------- problem constants (from reference) ----------------
#define N_NODES 40000
#define E_EDGES 640000
// S_IN=64, V_IN=32, S_MID=96, V_MID=128, S_OUT=96, V_OUT=32, R=64, FC_HID=64, W=224

typedef __attribute__((ext_vector_type(2))) float v2f;
typedef __attribute__((ext_vector_type(8))) float v8f;
typedef __attribute__((ext_vector_type(4))) unsigned int u32x4;
typedef __attribute__((ext_vector_type(4))) int i32x4;
typedef __attribute__((ext_vector_type(8))) int i32x8;

__device__ __forceinline__ float silu_f(float x) { return x / (1.f + __expf(-x)); }
__device__ __forceinline__ float sigm_f(float x) { return 1.f / (1.f + __expf(-x)); }

// Low 32 bits of a generic pointer to an LDS object == byte offset in LDS
// (flat LDS aperture keeps the offset in addr[31:0], ISA 10.2).
__device__ __forceinline__ unsigned lds_off(const void* p) {
    return (unsigned)(unsigned long long)p;
}

#if __has_builtin(__builtin_amdgcn_tensor_load_to_lds)
#define HAVE_TDM 1
// Issue one TDM load of a 2-D row-major tensor (tile == whole tensor) into LDS.
// D# layout per CDNA5 ISA ch.8: group0 = {flags, lds_addr, global_addr, type},
// group1 = {data_size, tensor_dim0/1, tile_dim0/1, dim0 stride}; groups 2/3 zero (2-D).
__device__ __forceinline__ void tdm_load_2d_f32(const void* gptr, unsigned ldsByteOff,
                                                unsigned dim0, unsigned dim1) {
    unsigned long long ga = (unsigned long long)gptr;
    u32x4 g0;
    g0.x = 0x1u;                                            // count=1, user mode
    g0.y = ldsByteOff;                                      // lds_addr (bytes)
    g0.z = (unsigned)(ga & 0xFFFFFFFFu);                    // global_addr[31:0]
    g0.w = (unsigned)((ga >> 32) & 0x01FFFFFFu) | (2u << 30); // addr[56:32] | type=2
    i32x8 g1;
    g1[0] = (int)(2u << 16);                                // data_size = 2 (4 bytes)
    g1[1] = (int)((dim0 & 0xFFFFu) << 16);                  // tensor_dim0[15:0] @ bits 63:48
    g1[2] = (int)(((dim0 >> 16) & 0xFFFFu) | ((dim1 & 0xFFFFu) << 16)); // dim0 hi | dim1 lo
    g1[3] = (int)(((dim1 >> 16) & 0xFFFFu) | ((dim0 & 0xFFFFu) << 16)); // dim1 hi | tile_dim0
    g1[4] = (int)(dim1 & 0xFFFFu);                          // tile_dim1 (tile_dim2 = 0)
    g1[5] = (int)dim0;                                      // tensor_dim0_stride low32 (= dim0)
    g1[6] = 0;                                              // stride hi / dim1_stride lo
    g1[7] = 0;
    i32x4 gz = {0, 0, 0, 0};
#if defined(__clang_major__) && __clang_major__ >= 23
    i32x8 gz8 = {0, 0, 0, 0, 0, 0, 0, 0};
    __builtin_amdgcn_tensor_load_to_lds(g0, g1, gz, gz, gz8, 0);
#else
    __builtin_amdgcn_tensor_load_to_lds(g0, g1, gz, gz, 0);
#endif
}
__device__ __forceinline__ void tdm_wait0() {
#if __has_builtin(__builtin_amdgcn_s_wait_tensorcnt)
    __builtin_amdgcn_s_wait_tensorcnt(0);
#else
    asm volatile("s_wait_tensorcnt 0x0" ::: "memory");
#endif
}
#else
#define HAVE_TDM 0
#endif

// ---------------- kernel 0: zero aggregation workspace ----------------
__global__ void zero_kernel(float* __restrict__ p, long n) {
    long i = (long)blockIdx.x * blockDim.x + threadIdx.x;
    if (i < n) p[i] = 0.f;
}

// ---------------- kernel 1: per-node h_s = s@lin1_ws*a/8, h_v = v@lin1_wv*a/sqrt(32) ----------------
__global__ void node_pre_kernel(const float* __restrict__ nf, const float* __restrict__ na,
                                const float* __restrict__ w_s, const float* __restrict__ w_v,
                                float* __restrict__ hs, float* __restrict__ hv) {
    __shared__ float sWs[64 * 64];
    __shared__ float sWv[32 * 32];
    for (int i = threadIdx.x; i < 64 * 64; i += blockDim.x) sWs[i] = w_s[i];
    for (int i = threadIdx.x; i < 32 * 32; i += blockDim.x) sWv[i] = w_v[i];
    __syncthreads();
    int n = blockIdx.x * blockDim.x + threadIdx.x;
    if (n >= N_NODES) return;
    const float a  = na[n];
    const float cs = a * 0.125f;                 // 1/sqrt(64)
    const float cv = a * 0.17677669529663687f;   // 1/sqrt(32)
    const float* row = nf + (long)n * 160;
    for (int d = 0; d < 64; ++d) {
        float acc = 0.f;
        for (int c = 0; c < 64; ++c) acc += row[c] * sWs[c * 64 + d];
        hs[(long)n * 64 + d] = acc * cs;
    }
    for (int d = 0; d < 32; ++d) {
        float a0 = 0.f, a1 = 0.f, a2 = 0.f;
        for (int c = 0; c < 32; ++c) {
            float w = sWv[c * 32 + d];
            a0 += row[64 + c * 3 + 0] * w;
            a1 += row[64 + c * 3 + 1] * w;
            a2 += row[64 + c * 3 + 2] * w;
        }
        hv[(long)n * 96 + d * 3 + 0] = a0 * cv;
        hv[(long)n * 96 + d * 3 + 1] = a1 * cv;
        hv[(long)n * 96 + d * 3 + 2] = a2 * cv;
    }
}

// ---------------- kernel 2: edge kernel (TDM weight staging + WMMA FC + atomic scatter) ----------------
// One wave = one 16-edge tile. 8 waves / 256-thread block. Grid = E/128 = 5000.
__global__ __launch_bounds__(256) void edge_kernel(
    const float* __restrict__ escal,   // (E,64)
    const int*   __restrict__ esrc,
    const int*   __restrict__ edst,
    const float* __restrict__ eattr,   // (E,4)
    const float* __restrict__ fc_w1,   // (64,64)
    const float* __restrict__ fc_w2,   // (64,224)
    const float* __restrict__ hs,      // (N,64)
    const float* __restrict__ hv,      // (N,96) = (N,32,3)
    float* __restrict__ agg_s,         // (N,96)
    float* __restrict__ agg_v)         // (N,384) = (N,128,3)
{
    __shared__ float sW1[64 * 64];      // 16KB  (TDM-filled)
    __shared__ float sW2[64 * 224];     // 56KB  (TDM-filled)
    __shared__ float sHid[8][16 * 64];  // 32KB: per-wave hid staging (C-layout -> A-layout)
    __shared__ int   sSrc[8][16];
    __shared__ int   sDst[8][16];
    __shared__ float sE0[8][16];
    __shared__ float sE1[8][16][3];

    const int wave = threadIdx.x >> 5;
    const int lane = threadIdx.x & 31;
    const int col  = lane & 15;   // N-column inside 16-tile / A-row (M)
    const int half = lane >> 4;   // which K/M half this lane holds
    const int kk   = half * 2;    // A/B fragment K sub-offset

    const long ebase = ((long)blockIdx.x * 8 + wave) * 16;

    // --- stage weights into LDS via Tensor Data Mover (wave 0 issues; TENSORcnt tracks) ---
#if HAVE_TDM
    if (threadIdx.x < 32) {
        tdm_load_2d_f32(fc_w1, lds_off(sW1), 64, 64);
        tdm_load_2d_f32(fc_w2, lds_off(sW2), 224, 64);
        tdm_wait0();
    }
#else
    for (int i = threadIdx.x; i < 64 * 64; i += 256) sW1[i] = fc_w1[i];
    for (int i = threadIdx.x; i < 64 * 224; i += 256) sW2[i] = fc_w2[i];
#endif

    if (lane < 16) {
        long e = ebase + lane;
        sSrc[wave][lane]   = esrc[e];
        sDst[wave][lane]   = edst[e];
        sE0[wave][lane]    = eattr[e * 4 + 0];
        sE1[wave][lane][0] = eattr[e * 4 + 1];
        sE1[wave][lane][1] = eattr[e * 4 + 2];
        sE1[wave][lane][2] = eattr[e * 4 + 3];
    }
    __syncthreads();

    // ---- stage 1: hid(16x64) = silu( X(16x64) @ W1(64x64) * 1/sqrt(64) ) ----
    const float* X = escal + ebase * 64;
    v8f acc[4] = {};
    for (int k = 0; k < 64; k += 4) {
        v2f a = *(const v2f*)(X + col * 64 + k + kk);   // global_load_b64
#pragma unroll
        for (int n = 0; n < 4; ++n) {
            v2f b;
            b.x = sW1[(k + kk    ) * 64 + n * 16 + col];
            b.y = sW1[(k + kk + 1) * 64 + n * 16 + col];
            acc[n] = __builtin_amdgcn_wmma_f32_16x16x4_f32(
                false, a, false, b, (short)0, acc[n], false, false);
        }
    }
    // silu elementwise in C-layout, then stage to LDS row-major for re-fragmentation
#pragma unroll
    for (int n = 0; n < 4; ++n) {
#pragma unroll
        for (int r = 0; r < 8; ++r) {
            int row = r + 8 * half;  // C/D layout: VGPR r holds M=r (lanes 0-15), M=r+8 (16-31)
            sHid[wave][row * 64 + n * 16 + col] = silu_f(acc[n][r] * 0.125f);
        }
    }
    __syncthreads();

    // ---- stage 2: w(16x224) = hid(16x64) @ W2(64x224) * 1/sqrt(64) ----
    v8f wacc[14] = {};
    for (int k = 0; k < 64; k += 4) {
        v2f a;
        a.x = sHid[wave][col * 64 + k + kk];
        a.y = sHid[wave][col * 64 + k + kk + 1];
#pragma unroll
        for (int n = 0; n < 14; ++n) {
            v2f b;
            b.x = sW2[(k + kk    ) * 224 + n * 16 + col];  // ds_load_2addr (row stride 224 dw)
            b.y = sW2[(k + kk + 1) * 224 + n * 16 + col];
            wacc[n] = __builtin_amdgcn_wmma_f32_16x16x4_f32(
                false, a, false, b, (short)0, wacc[n], false, false);
        }
    }

    // ---- stage 3: messages + segment-sum via f32 atomics (agg buffers are L2-resident) ----
    const float scale_w = 0.125f;               // 1/sqrt(FC_HID)
    const float inv_nn  = 0.25f;                // 1/sqrt(16) fused here
    const float c_s2    = 0.57735026918962584f; // 1/sqrt(3)
    const float c_v3    = 0.70710678118654746f; // 1/sqrt(2)

#pragma unroll
    for (int r = 0; r < 8; ++r) {
        const int m   = r + 8 * half;           // edge row this lane's acc element r belongs to
        const int src = sSrc[wave][m];
        const int dst = sDst[wave][m];
        const float e0v = sE0[wave][m];
        const float e1x = sE1[wave][m][0];
        const float e1y = sE1[wave][m][1];
        const float e1z = sE1[wave][m][2];
        const float* hsrow = hs + (long)src * 64;
        const float* hvrow = hv + (long)src * 96;
        float* as = agg_s + (long)dst * 96;
        float* av = agg_v + (long)dst * 384;

        const float es[4] = { hsrow[col], hsrow[col + 16], hsrow[col + 32], hsrow[col + 48] };
        const float evA0 = hvrow[col * 3 + 0], evA1 = hvrow[col * 3 + 1], evA2 = hvrow[col * 3 + 2];
        const float evB0 = hvrow[(col + 16) * 3 + 0], evB1 = hvrow[(col + 16) * 3 + 1], evB2 = hvrow[(col + 16) * 3 + 2];

        // m_s1 = w1 * es * e0   (features 0..63 of agg_s)
#pragma unroll
        for (int n = 0; n < 4; ++n) {
            float v = wacc[n][r] * scale_w * es[n] * e0v * inv_nn;
            unsafeAtomicAdd(as + n * 16 + col, v);
        }
        // m_v1 = (w2*es) outer e1  (channels 0..63 of agg_v)
#pragma unroll
        for (int n = 0; n < 4; ++n) {
            float v = wacc[4 + n][r] * scale_w * es[n] * inv_nn;
            float* p = av + (n * 16 + col) * 3;
            unsafeAtomicAdd(p + 0, v * e1x);
            unsafeAtomicAdd(p + 1, v * e1y);
            unsafeAtomicAdd(p + 2, v * e1z);
        }
        // m_v2 = w3 * ev * e0  (channels 64..95)
        {
            float vA = wacc[8][r] * scale_w * e0v * inv_nn;
            float* p = av + (64 + col) * 3;
            unsafeAtomicAdd(p + 0, vA * evA0);
            unsafeAtomicAdd(p + 1, vA * evA1);
            unsafeAtomicAdd(p + 2, vA * evA2);
            float vB = wacc[9][r] * scale_w * e0v * inv_nn;
            float* q = av + (80 + col) * 3;
            unsafeAtomicAdd(q + 0, vB * evB0);
            unsafeAtomicAdd(q + 1, vB * evB1);
            unsafeAtomicAdd(q + 2, vB * evB2);
        }
        // m_s2 = w4 * dot(ev, e1) / sqrt(3)  (features 64..95 of agg_s)
        {
            float dA = evA0 * e1x + evA1 * e1y + evA2 * e1z;
            float dB = evB0 * e1x + evB1 * e1y + evB2 * e1z;
            unsafeAtomicAdd(as + 64 + col, wacc[10][r] * scale_w * dA * c_s2 * inv_nn);
            unsafeAtomicAdd(as + 80 + col, wacc[11][r] * scale_w * dB * c_s2 * inv_nn);
        }
        // m_v3 = w5 * cross(ev, e1) / sqrt(2)  (channels 96..127)
        {
            float cxA = evA1 * e1z - evA2 * e1y;
            float cyA = evA2 * e1x - evA0 * e1z;
            float czA = evA0 * e1y - evA1 * e1x;
            float vA = wacc[12][r] * scale_w * c_v3 * inv_nn;
            float* p = av + (96 + col) * 3;
            unsafeAtomicAdd(p + 0, vA * cxA);
            unsafeAtomicAdd(p + 1, vA * cyA);
            unsafeAtomicAdd(p + 2, vA * czA);
            float cxB = evB1 * e1z - evB2 * e1y;
            float cyB = evB2 * e1x - evB0 * e1z;
            float czB = evB0 * e1y - evB1 * e1x;
            float vB = wacc[13][r] * scale_w * c_v3 * inv_nn;
            float* q = av + (112 + col) * 3;
            unsafeAtomicAdd(q + 0, vB * cxB);
            unsafeAtomicAdd(q + 1, vB * cyB);
            unsafeAtomicAdd(q + 2, vB * czB);
        }
    }
}

// ---------------- kernel 3: skip-connection + output linears + gating ----------------
__global__ void node_out_kernel(const float* __restrict__ nf, const float* __restrict__ na,
                                const float* __restrict__ agg_s, const float* __restrict__ agg_v,
                                const float* __restrict__ sc_ws, const float* __restrict__ sc_wv,
                                const float* __restrict__ lin2_ws, const float* __restrict__ lin2_wv,
                                float* __restrict__ out) {
    int n = blockIdx.x * blockDim.x + threadIdx.x;
    if (n >= N_NODES) return;
    const float a = na[n];
    const float k_sc_s = a * 0.125f;                // 1/sqrt(64)
    const float k_sc_v = a * 0.17677669529663687f;  // 1/sqrt(32)
    const float k_o_s  = a * 0.10206207261596575f;  // 1/sqrt(96)
    const float k_o_v  = a * 0.08838834764831845f;  // 1/sqrt(128)
    const float* row = nf + (long)n * 160;
    const float* as  = agg_s + (long)n * 96;
    const float* av  = agg_v + (long)n * 384;

    // scalar outputs 0..63
    for (int d = 0; d < 64; ++d) {
        float o = 0.f, sc = 0.f;
        for (int c = 0; c < 96; ++c) o  += as[c]  * lin2_ws[c * 96 + d];
        for (int c = 0; c < 64; ++c) sc += row[c] * sc_ws[c * 96 + d];
        out[(long)n * 160 + d] = silu_f(sc * k_sc_s + o * k_o_s);
    }
    // gated vector outputs
    for (int d = 0; d < 32; ++d) {
        float o = 0.f, sc = 0.f;
        for (int c = 0; c < 96; ++c) o  += as[c]  * lin2_ws[c * 96 + 64 + d];
        for (int c = 0; c < 64; ++c) sc += row[c] * sc_ws[c * 96 + 64 + d];
        float gate = sigm_f(sc * k_sc_s + o * k_o_s);

        float o0 = 0.f, o1 = 0.f, o2 = 0.f, s0 = 0.f, s1 = 0.f, s2 = 0.f;
        for (int c = 0; c < 128; ++c) {
            float w = lin2_wv[c * 32 + d];
            o0 += av[c * 3 + 0] * w;
            o1 += av[c * 3 + 1] * w;
            o2 += av[c * 3 + 2] * w;
        }
        for (int c = 0; c < 32; ++c) {
            float w = sc_wv[c * 32 + d];
            s0 += row[64 + c * 3 + 0] * w;
            s1 += row[64 + c * 3 + 1] * w;
            s2 += row[64 + c * 3 + 2] * w;
        }
        out[(long)n * 160 + 64 + d * 3 + 0] = gate * (s0 * k_sc_v + o0 * k_o_v);
        out[(long)n * 160 + 64 + d * 3 + 1] = gate * (s1 * k_sc_v + o1 * k_o_v);
        out[(long)n * 160 + 64 + d * 3 + 2] = gate * (s2 * k_sc_v + o2 * k_o_v);
    }
}

// ---------------- launch ----------------
extern "C" void kernel_launch(void* const* d_in, const int* in_sizes, int n_in,
                              void* d_out, int out_size, void* d_ws, size_t ws_size,
                              hipStream_t stream) {
    const float* nf      = (const float*)d_in[0];
    const float* na      = (const float*)d_in[1];
    const int*   esrc    = (const int*)d_in[2];
    const int*   edst    = (const int*)d_in[3];
    const float* eattr   = (const float*)d_in[4];
    const float* escal   = (const float*)d_in[5];
    const float* lin1_ws = (const float*)d_in[6];
    const float* lin1_wv = (const float*)d_in[7];
    const float* fc_w1   = (const float*)d_in[8];
    const float* fc_w2   = (const float*)d_in[9];
    const float* sc_ws   = (const float*)d_in[10];
    const float* sc_wv   = (const float*)d_in[11];
    const float* lin2_ws = (const float*)d_in[12];
    const float* lin2_wv = (const float*)d_in[13];
    float* out = (float*)d_out;

    float* ws    = (float*)d_ws;
    float* hs    = ws;                             // N*64
    float* hv    = hs    + (size_t)N_NODES * 64;   // N*96
    float* agg_s = hv    + (size_t)N_NODES * 96;   // N*96
    float* agg_v = agg_s + (size_t)N_NODES * 96;   // N*384
    const long aggTot = (long)N_NODES * (96 + 384);

    zero_kernel<<<(int)((aggTot + 511) / 512), 512, 0, stream>>>(agg_s, aggTot);
    node_pre_kernel<<<(N_NODES + 127) / 128, 128, 0, stream>>>(nf, na, lin1_ws, lin1_wv, hs, hv);
    edge_kernel<<<E_EDGES / 128, 256, 0, stream>>>(escal, esrc, edst, eattr, fc_w1, fc_w2,
                                                   hs, hv, agg_s, agg_v);
    node_out_kernel<<<(N_NODES + 127) / 128, 128, 0, stream>>>(nf, na, agg_s, agg_v,
                                                               sc_ws, sc_wv, lin2_ws, lin2_wv, out);
}